// myDC_15710990369562
// MI455X (gfx1250) — compile-verified
//
#include <hip/hip_runtime.h>

typedef float v2f __attribute__((ext_vector_type(2)));
typedef float v8f __attribute__((ext_vector_type(8)));

#define NXY     65536   // 256*256
#define NB      4
#define NC      12
#define NIMG    48      // NB*NC
#define ITERS   10
#define TOL     1e-8f

// ---------------------------------------------------------------------------
// f32 WMMA 16x16x4:  D(16x16 f32) = A(16x4 f32) * B(4x16 f32) + C
// ---------------------------------------------------------------------------
__device__ __forceinline__ v8f wmma4(v2f a, v2f b, v8f c) {
  return __builtin_amdgcn_wmma_f32_16x16x4_f32(false, a, false, b, (short)0, c,
                                               false, false);
}

// ---------------------------------------------------------------------------
// Build centered-DFT matrices.  fftshift*F*fftshift == checkerboard(F):
//   W [k,n] = (-1)^(k+n) * exp(-2*pi*i*k*n/256)
//   Wi[k,n] = (-1)^(k+n) * exp(+2*pi*i*k*n/256) / 256
// ---------------------------------------------------------------------------
__global__ void build_w_k(float* __restrict__ wf_re, float* __restrict__ wf_im,
                          float* __restrict__ wi_re, float* __restrict__ wi_im) {
  int idx = blockIdx.x * blockDim.x + threadIdx.x;
  if (idx >= NXY) return;
  int k = idx >> 8, n = idx & 255;
  int t = (k * n) & 255;                       // exact periodic reduction
  double th = (double)t * 0.024543692606170259; // 2*pi/256
  float cs = (float)cos(th);
  float sn = (float)sin(th);
  float s  = ((k + n) & 1) ? -1.0f : 1.0f;
  wf_re[idx] =  s * cs;
  wf_im[idx] = -s * sn;
  wi_re[idx] =  s * cs * (1.0f / 256.0f);
  wi_im[idx] =  s * sn * (1.0f / 256.0f);
}

// ---------------------------------------------------------------------------
// Complex GEMM (256x256x256) per image:  C = A * B, optional mask epilogue.
// sA/sB = per-image element stride (0 => shared matrix, e.g. W).
// Grid: (4, 2, nImages), 256 threads (8 waves); wave owns a 32x32 tile.
// ---------------------------------------------------------------------------
__global__ __launch_bounds__(256) void cgemm_wmma_k(
    const float* __restrict__ Are, const float* __restrict__ Aim, int sA,
    const float* __restrict__ Bre, const float* __restrict__ Bim, int sB,
    float* __restrict__ Cre, float* __restrict__ Cim,
    const float* __restrict__ mask) {
  const int img  = blockIdx.z;
  const int lane = threadIdx.x & 31;
  const int wid  = threadIdx.x >> 5;
  const int rowBase = blockIdx.x * 64  + (wid >> 2) * 32;
  const int colBase = blockIdx.y * 128 + (wid & 3) * 32;

  const float* are = Are + (size_t)img * sA;
  const float* aim = Aim + (size_t)img * sA;
  const float* bre = Bre + (size_t)img * sB;
  const float* bim = Bim + (size_t)img * sB;

  const int l15 = lane & 15;
  const int kHi = (lane >> 4) << 1;     // lane half selects K pair {0,1}/{2,3}
  const int r0 = rowBase + l15;
  const int r1 = r0 + 16;
  const int c0 = colBase + l15;
  const int c1 = c0 + 16;

  v8f cr00 = {}, cr01 = {}, cr10 = {}, cr11 = {};
  v8f ci00 = {}, ci01 = {}, ci10 = {}, ci11 = {};

#pragma unroll 4
  for (int k = 0; k < 256; k += 4) {
    const int kc = k + kHi;
    // A frags: lane = M row, two consecutive K per lane (8B load)
    v2f ar0 = *(const v2f*)(are + (size_t)r0 * 256 + kc);
    v2f ar1 = *(const v2f*)(are + (size_t)r1 * 256 + kc);
    v2f ai0 = *(const v2f*)(aim + (size_t)r0 * 256 + kc);
    v2f ai1 = *(const v2f*)(aim + (size_t)r1 * 256 + kc);
    v2f nai0 = -ai0;                    // f32 WMMA has no A/B NEG modifier
    v2f nai1 = -ai1;
    // B frags: lane = N col, VGPR index = consecutive K within lane-half
    const float* pbr = bre + (size_t)kc * 256;
    const float* pbi = bim + (size_t)kc * 256;
    v2f br0 = {pbr[c0], pbr[c0 + 256]};
    v2f br1 = {pbr[c1], pbr[c1 + 256]};
    v2f bi0 = {pbi[c0], pbi[c0 + 256]};
    v2f bi1 = {pbi[c1], pbi[c1 + 256]};

    cr00 = wmma4(ar0,  br0, cr00);  cr00 = wmma4(nai0, bi0, cr00);
    ci00 = wmma4(ar0,  bi0, ci00);  ci00 = wmma4(ai0,  br0, ci00);

    cr01 = wmma4(ar0,  br1, cr01);  cr01 = wmma4(nai0, bi1, cr01);
    ci01 = wmma4(ar0,  bi1, ci01);  ci01 = wmma4(ai0,  br1, ci01);

    cr10 = wmma4(ar1,  br0, cr10);  cr10 = wmma4(nai1, bi0, cr10);
    ci10 = wmma4(ar1,  bi0, ci10);  ci10 = wmma4(ai1,  br0, ci10);

    cr11 = wmma4(ar1,  br1, cr11);  cr11 = wmma4(nai1, bi1, cr11);
    ci11 = wmma4(ar1,  bi1, ci11);  ci11 = wmma4(ai1,  br1, ci11);
  }

  float* cre = Cre + (size_t)img * NXY;
  float* cim = Cim + (size_t)img * NXY;
  const float* mb = mask ? (mask + (size_t)(img / NC) * NXY) : nullptr;
  const int mOff = (lane >> 4) ? 8 : 0;   // C layout: lane-half = M half

  auto store_tile = [&](v8f cr, v8f ci, int rb, int cb) {
#pragma unroll
    for (int v = 0; v < 8; ++v) {
      const size_t o = (size_t)(rb + v) * 256 + cb;
      const float m = mb ? mb[o] : 1.0f;
      cre[o] = cr[v] * m;
      cim[o] = ci[v] * m;
    }
  };
  store_tile(cr00, ci00, rowBase      + mOff, colBase      + l15);
  store_tile(cr01, ci01, rowBase      + mOff, colBase + 16 + l15);
  store_tile(cr10, ci10, rowBase + 16 + mOff, colBase      + l15);
  store_tile(cr11, ci11, rowBase + 16 + mOff, colBase + 16 + l15);
}

// ---------------------------------------------------------------------------
// Elementwise / reduction kernels
// ---------------------------------------------------------------------------
__global__ void init_k(const float* __restrict__ rhs,
                       float* __restrict__ x_re, float* __restrict__ x_im,
                       float* __restrict__ r_re, float* __restrict__ r_im,
                       float* __restrict__ p_re, float* __restrict__ p_im) {
  int t = blockIdx.x * blockDim.x + threadIdx.x;   // NB*NXY
  float re = rhs[2 * t], im = rhs[2 * t + 1];
  r_re[t] = re; r_im[t] = im;
  p_re[t] = re; p_im[t] = im;
  x_re[t] = 0.0f; x_im[t] = 0.0f;
}

__global__ void coilmul_k(const float* __restrict__ csm_r,
                          const float* __restrict__ csm_i,
                          const float* __restrict__ p_re,
                          const float* __restrict__ p_im,
                          float* __restrict__ co_re,
                          float* __restrict__ co_im) {
  int t = blockIdx.x * blockDim.x + threadIdx.x;   // NIMG*NXY
  int g = t >> 16;
  int b = g / NC;
  int pb = (b << 16) + (t & 65535);
  float sr = csm_r[t], si = csm_i[t];
  float pr = p_re[pb], pi = p_im[pb];
  co_re[t] = sr * pr - si * pi;
  co_im[t] = sr * pi + si * pr;
}

__global__ void combine_k(const float* __restrict__ ci_re,
                          const float* __restrict__ ci_im,
                          const float* __restrict__ csm_r,
                          const float* __restrict__ csm_i,
                          const float* __restrict__ p_re,
                          const float* __restrict__ p_im,
                          const float* __restrict__ lam1,
                          const float* __restrict__ lam2,
                          float* __restrict__ Ap_re,
                          float* __restrict__ Ap_im) {
  int t = blockIdx.x * blockDim.x + threadIdx.x;   // NB*NXY
  int b = t >> 16;
  int pix = t & 65535;
  float lam = lam1[0] + lam2[0];
  float ar = 0.0f, ai = 0.0f;
  size_t base = ((size_t)b * NC) << 16;
#pragma unroll 4
  for (int c = 0; c < NC; ++c) {
    size_t o = base + ((size_t)c << 16) + pix;
    float cr = ci_re[o], cii = ci_im[o];
    float sr = csm_r[o], si = csm_i[o];
    ar += cr * sr + cii * si;      // Re(ci * conj(csm))
    ai += cii * sr - cr * si;      // Im(ci * conj(csm))
  }
  Ap_re[t] = ar + lam * p_re[t];
  Ap_im[t] = ai + lam * p_im[t];
}

__global__ void dot_partial_k(const float* __restrict__ a_re,
                              const float* __restrict__ a_im,
                              const float* __restrict__ b_re,
                              const float* __restrict__ b_im,
                              float* __restrict__ partial) {
  __shared__ float s[256];
  int b = blockIdx.y;
  int t = (b << 16) + blockIdx.x * 256 + threadIdx.x;
  s[threadIdx.x] = a_re[t] * b_re[t] + a_im[t] * b_im[t];
  __syncthreads();
  for (int o = 128; o > 0; o >>= 1) {
    if (threadIdx.x < o) s[threadIdx.x] += s[threadIdx.x + o];
    __syncthreads();
  }
  if (threadIdx.x == 0) partial[b * 256 + blockIdx.x] = s[0];
}

__global__ void reduce_final_k(const float* __restrict__ partial,
                               float* __restrict__ out) {
  __shared__ float s[256];
  int b = blockIdx.x;
  s[threadIdx.x] = partial[b * 256 + threadIdx.x];
  __syncthreads();
  for (int o = 128; o > 0; o >>= 1) {
    if (threadIdx.x < o) s[threadIdx.x] += s[threadIdx.x + o];
    __syncthreads();
  }
  if (threadIdx.x == 0) out[b] = s[0];
}

__global__ void update_xr_k(float* __restrict__ x_re, float* __restrict__ x_im,
                            float* __restrict__ r_re, float* __restrict__ r_im,
                            const float* __restrict__ p_re,
                            const float* __restrict__ p_im,
                            const float* __restrict__ Ap_re,
                            const float* __restrict__ Ap_im,
                            const float* __restrict__ rTr,
                            const float* __restrict__ pAp,
                            float* __restrict__ partialN) {
  __shared__ float s[256];
  int b = blockIdx.y;
  int t = (b << 16) + blockIdx.x * 256 + threadIdx.x;
  float rt = rTr[b];
  bool act = rt > TOL;
  float alpha = rt / pAp[b];
  float rr, ri;
  if (act) {
    x_re[t] += alpha * p_re[t];
    x_im[t] += alpha * p_im[t];
    rr = r_re[t] - alpha * Ap_re[t];
    ri = r_im[t] - alpha * Ap_im[t];
    r_re[t] = rr; r_im[t] = ri;
  } else {
    rr = r_re[t]; ri = r_im[t];
  }
  s[threadIdx.x] = rr * rr + ri * ri;
  __syncthreads();
  for (int o = 128; o > 0; o >>= 1) {
    if (threadIdx.x < o) s[threadIdx.x] += s[threadIdx.x + o];
    __syncthreads();
  }
  if (threadIdx.x == 0) partialN[b * 256 + blockIdx.x] = s[0];
}

__global__ void scalar_k(const float* __restrict__ partialN,
                         float* __restrict__ rTr, float* __restrict__ beta,
                         float* __restrict__ actF) {
  __shared__ float s[256];
  int b = blockIdx.x;
  s[threadIdx.x] = partialN[b * 256 + threadIdx.x];
  __syncthreads();
  for (int o = 128; o > 0; o >>= 1) {
    if (threadIdx.x < o) s[threadIdx.x] += s[threadIdx.x + o];
    __syncthreads();
  }
  if (threadIdx.x == 0) {
    float rtn = s[0];
    float rt  = rTr[b];
    bool act  = rt > TOL;
    beta[b] = rtn / rt;
    actF[b] = act ? 1.0f : 0.0f;
    if (act) rTr[b] = rtn;
  }
}

__global__ void update_p_k(float* __restrict__ p_re, float* __restrict__ p_im,
                           const float* __restrict__ r_re,
                           const float* __restrict__ r_im,
                           const float* __restrict__ beta,
                           const float* __restrict__ actF) {
  int b = blockIdx.y;
  int t = (b << 16) + blockIdx.x * 256 + threadIdx.x;
  if (actF[b] != 0.0f) {
    float bb = beta[b];
    p_re[t] = r_re[t] + bb * p_re[t];
    p_im[t] = r_im[t] + bb * p_im[t];
  }
}

__global__ void writeout_k(const float* __restrict__ x_re,
                           const float* __restrict__ x_im,
                           float* __restrict__ out) {
  int t = blockIdx.x * blockDim.x + threadIdx.x;   // NB*NXY
  out[2 * t]     = x_re[t];
  out[2 * t + 1] = x_im[t];
}

// ---------------------------------------------------------------------------
extern "C" void kernel_launch(void* const* d_in, const int* in_sizes, int n_in,
                              void* d_out, int out_size, void* d_ws,
                              size_t ws_size, hipStream_t stream) {
  (void)in_sizes; (void)n_in; (void)out_size; (void)ws_size;
  const float* rhs   = (const float*)d_in[0];
  const float* csm_r = (const float*)d_in[1];
  const float* csm_i = (const float*)d_in[2];
  const float* mask  = (const float*)d_in[3];
  const float* lam1  = (const float*)d_in[4];
  const float* lam2  = (const float*)d_in[5];
  float* out = (float*)d_out;

  // Workspace carve-up (floats). Total ~= 60 MB.
  float* ws = (float*)d_ws;
  size_t o = 0;
  float* WF_RE = ws + o; o += NXY;
  float* WF_IM = ws + o; o += NXY;
  float* WI_RE = ws + o; o += NXY;
  float* WI_IM = ws + o; o += NXY;
  float* x_re  = ws + o; o += (size_t)NB * NXY;
  float* x_im  = ws + o; o += (size_t)NB * NXY;
  float* r_re  = ws + o; o += (size_t)NB * NXY;
  float* r_im  = ws + o; o += (size_t)NB * NXY;
  float* p_re  = ws + o; o += (size_t)NB * NXY;
  float* p_im  = ws + o; o += (size_t)NB * NXY;
  float* Ap_re = ws + o; o += (size_t)NB * NXY;
  float* Ap_im = ws + o; o += (size_t)NB * NXY;
  float* bufA_re = ws + o; o += (size_t)NIMG * NXY;
  float* bufA_im = ws + o; o += (size_t)NIMG * NXY;
  float* bufB_re = ws + o; o += (size_t)NIMG * NXY;
  float* bufB_im = ws + o; o += (size_t)NIMG * NXY;
  float* rTr   = ws + o; o += NB;
  float* pAp   = ws + o; o += NB;
  float* beta  = ws + o; o += NB;
  float* actF  = ws + o; o += NB;
  float* partial  = ws + o; o += (size_t)NB * 256;
  float* partialN = ws + o; o += (size_t)NB * 256;

  const dim3 gGemm(4, 2, NIMG);
  const dim3 gBatch(256, NB);

  build_w_k<<<NXY / 256, 256, 0, stream>>>(WF_RE, WF_IM, WI_RE, WI_IM);
  init_k<<<NB * NXY / 256, 256, 0, stream>>>(rhs, x_re, x_im, r_re, r_im,
                                             p_re, p_im);
  dot_partial_k<<<gBatch, 256, 0, stream>>>(r_re, r_im, r_re, r_im, partial);
  reduce_final_k<<<NB, 256, 0, stream>>>(partial, rTr);

  for (int it = 0; it < ITERS; ++it) {
    // coil = csm * p
    coilmul_k<<<NIMG * NXY / 256, 256, 0, stream>>>(csm_r, csm_i, p_re, p_im,
                                                    bufA_re, bufA_im);
    // T = W * coil
    cgemm_wmma_k<<<gGemm, 256, 0, stream>>>(WF_RE, WF_IM, 0, bufA_re, bufA_im,
                                            NXY, bufB_re, bufB_im, nullptr);
    // K = (T * W) .* mask
    cgemm_wmma_k<<<gGemm, 256, 0, stream>>>(bufB_re, bufB_im, NXY, WF_RE,
                                            WF_IM, 0, bufA_re, bufA_im, mask);
    // T2 = Wi * K
    cgemm_wmma_k<<<gGemm, 256, 0, stream>>>(WI_RE, WI_IM, 0, bufA_re, bufA_im,
                                            NXY, bufB_re, bufB_im, nullptr);
    // ci = T2 * Wi
    cgemm_wmma_k<<<gGemm, 256, 0, stream>>>(bufB_re, bufB_im, NXY, WI_RE,
                                            WI_IM, 0, bufA_re, bufA_im,
                                            nullptr);
    // Ap = sum_c ci*conj(csm) + lam*p
    combine_k<<<NB * NXY / 256, 256, 0, stream>>>(bufA_re, bufA_im, csm_r,
                                                  csm_i, p_re, p_im, lam1,
                                                  lam2, Ap_re, Ap_im);
    // pAp, CG updates
    dot_partial_k<<<gBatch, 256, 0, stream>>>(p_re, p_im, Ap_re, Ap_im,
                                              partial);
    reduce_final_k<<<NB, 256, 0, stream>>>(partial, pAp);
    update_xr_k<<<gBatch, 256, 0, stream>>>(x_re, x_im, r_re, r_im, p_re,
                                            p_im, Ap_re, Ap_im, rTr, pAp,
                                            partialN);
    scalar_k<<<NB, 256, 0, stream>>>(partialN, rTr, beta, actF);
    update_p_k<<<gBatch, 256, 0, stream>>>(p_re, p_im, r_re, r_im, beta, actF);
  }

  writeout_k<<<NB * NXY / 256, 256, 0, stream>>>(x_re, x_im, out);
}